// RoIPooling_74715251081429
// MI455X (gfx1250) — compile-verified
//
#include <hip/hip_runtime.h>
#include <cstdint>
#include <math.h>

// RoIPool, MI455X (gfx1250). Bandwidth/latency-bound gather-max:
//   - feature map (10.2 MB) is L2-resident (192 MB L2) -> direct global loads
//   - ROI descriptor broadcast via CDNA5 async global->LDS DMA + ASYNCcnt
//   - global_prefetch_b8 to warm the cache ahead of the bin max loop

#define PH 7
#define PW 7
#define SCALE_ 0.0625f
#define C_ 256
#define H_ 50
#define W_ 50
#define OUT_PER_ROI (C_ * PH * PW)                       // 12544
#define THREADS 256
#define BLOCKS_PER_ROI ((OUT_PER_ROI + THREADS - 1) / THREADS)  // 49

__global__ __launch_bounds__(THREADS)
void roipool_gfx1250_kernel(const float* __restrict__ features,
                            const float* __restrict__ rois,
                            float* __restrict__ out)
{
    __shared__ float sroi[8];

    const int r = blockIdx.y;

    // ---- Stage the 5-float ROI descriptor into LDS with the CDNA5 async
    // ---- global->LDS path. threadIdx.x < 32 is exactly wave 0 (wave32),
    // ---- so the s_wait_asynccnt sits in a wave-uniform branch.
    if (threadIdx.x < 32u) {
        if (threadIdx.x < 5u) {
            const float* gp = rois + (size_t)r * 5 + threadIdx.x;
            unsigned lds_addr = (unsigned)(uintptr_t)(&sroi[threadIdx.x]);
            asm volatile("global_load_async_to_lds_b32 %0, %1, off"
                         :: "v"(lds_addr), "v"(gp)
                         : "memory");
        }
        asm volatile("s_wait_asynccnt 0" ::: "memory");
    }
    __syncthreads();

    const int base = blockIdx.x * THREADS + (int)threadIdx.x;
    if (base >= OUT_PER_ROI) return;

    const int c  = base / (PH * PW);
    const int o  = base % (PH * PW);
    const int ph = o / PW;
    const int pw = o % PW;

    // ---- Bin geometry: bit-exact replay of the reference math.
    // jnp.round == round-half-even == v_rndne_f32 (rintf under default mode).
    const int   bidx = (int)sroi[0];
    const float x1 = rintf(sroi[1] * SCALE_);
    const float y1 = rintf(sroi[2] * SCALE_);
    const float x2 = rintf(sroi[3] * SCALE_);
    const float y2 = rintf(sroi[4] * SCALE_);

    const float roi_w = fmaxf(x2 - x1 + 1.0f, 1.0f);
    const float roi_h = fmaxf(y2 - y1 + 1.0f, 1.0f);
    const float bsw = roi_w / (float)PW;   // true IEEE divide, matches jnp
    const float bsh = roi_h / (float)PH;

    // floor(j*bs) + start, clip to [0, L], then int cast — same op order as ref
    const int ws = (int)fminf(fmaxf(floorf((float)pw       * bsw) + x1, 0.0f), (float)W_);
    const int we = (int)fminf(fmaxf(ceilf ((float)(pw + 1) * bsw) + x1, 0.0f), (float)W_);
    const int hs = (int)fminf(fmaxf(floorf((float)ph       * bsh) + y1, 0.0f), (float)H_);
    const int he = (int)fminf(fmaxf(ceilf ((float)(ph + 1) * bsh) + y1, 0.0f), (float)H_);

    const float* f = features + ((size_t)bidx * C_ + c) * (size_t)(H_ * W_);

    // Warm WGP$/L2 for this thread's bin region (emits global_prefetch_b8).
    __builtin_prefetch(f + hs * W_ + ws, 0, 0);

    float m = -INFINITY;
    for (int h = hs; h < he; ++h) {
        const float* row = f + h * W_;
        for (int w = ws; w < we; ++w) {
            m = fmaxf(m, row[w]);
        }
    }

    // Empty bin (loop never ran) -> 0, matching the reference's neg-inf fill.
    out[(size_t)r * OUT_PER_ROI + base] = (m == -INFINITY) ? 0.0f : m;
}

extern "C" void kernel_launch(void* const* d_in, const int* in_sizes, int n_in,
                              void* d_out, int out_size, void* d_ws, size_t ws_size,
                              hipStream_t stream)
{
    const float* features = (const float*)d_in[0];   // [N, 256, 50, 50] f32
    const float* rois     = (const float*)d_in[1];   // [R, 5]           f32
    float*       out      = (float*)d_out;           // [R, 256, 7, 7]   f32

    const int R = in_sizes[1] / 5;

    dim3 grid(BLOCKS_PER_ROI, R, 1);
    roipool_gfx1250_kernel<<<grid, THREADS, 0, stream>>>(features, rois, out);
}